// RetNetRelPosDeform2D_4131758539024
// MI455X (gfx1250) — compile-verified
//
#include <hip/hip_runtime.h>
#include <hip/hip_bf16.h>

typedef __attribute__((ext_vector_type(2))) float v2f;
typedef __attribute__((ext_vector_type(8))) float v8f;

#define NB 8
#define NH 8
#define N1 32
#define N2 32
#define SLEN 1024
#define DHEAD 64
#define ROWS 16
#define LSTRIDE 1032  // 1024 + 8 floats of pad to spread LDS banks across rows

// One block handles one (b,h) pair and a strip of 16 query rows.
// grid = (B*H) * (SLEN/16) = 64 * 64 = 4096 blocks, 256 threads (8 waves).
__global__ __launch_bounds__(256)
void deform_strip_kernel(const float* __restrict__ offsets,
                         const float* __restrict__ angle,
                         const float* __restrict__ decay,
                         float* __restrict__ sin_q,
                         float* __restrict__ cos_q,
                         float* __restrict__ mask_out)
{
    __shared__ float smem[ROWS * LSTRIDE];   // 16 x 1024 f32 mask tile (padded)
    __shared__ float s_i1[ROWS];
    __shared__ float s_i2[ROWS];
    __shared__ float s_idxq[ROWS];
    __shared__ float s_part[8][ROWS];        // per-wave partial row sums
    __shared__ float s_scale[ROWS];

    const int tid   = threadIdx.x;
    const int bid   = blockIdx.x;
    const int bh    = bid >> 6;              // 0..63 = b*8+h
    const int strip = bid & 63;              // 0..63
    const int qbase = strip * ROWS;
    const float dec = decay[bh & (NH - 1)];  // uniform scalar load

    // deformed query indices for the 16 rows of this strip
    if (tid < ROWS) {
        const int q = qbase + tid;
        const float off0 = offsets[((size_t)bh * SLEN + q) * 2 + 0];
        const float off1 = offsets[((size_t)bh * SLEN + q) * 2 + 1];
        const float i1 = (float)(q >> 5) + off0;        // i + off0
        const float i2 = (float)(q & (N2 - 1)) + off1;  // j + off1
        s_i1[tid] = i1;
        s_i2[tid] = i2;
        s_idxq[tid] = i1 + i2;
    }
    __syncthreads();

    // --- mask tile: iteration k fills row k; each thread does 4 columns ---
    const int c0 = tid * 4;                      // key position base, 0..1020
    const float kif = (float)(c0 >> 5);          // key i' (same for all 4 cols)
    const float kj0 = (float)(c0 & 31);          // key j' base
    for (int k = 0; k < ROWS; ++k) {
        const float i1 = s_i1[k];
        const float i2 = s_i2[k];
        const float d1 = fabsf(i1 - kif);
        float4 v;
        float e;
        e = __expf((d1 + fabsf(i2 - (kj0 + 0.0f))) * dec); v.x = (e == e) ? e : 0.0f;
        e = __expf((d1 + fabsf(i2 - (kj0 + 1.0f))) * dec); v.y = (e == e) ? e : 0.0f;
        e = __expf((d1 + fabsf(i2 - (kj0 + 2.0f))) * dec); v.z = (e == e) ? e : 0.0f;
        e = __expf((d1 + fabsf(i2 - (kj0 + 3.0f))) * dec); v.w = (e == e) ? e : 0.0f;
        *(float4*)&smem[k * LSTRIDE + c0] = v;
    }
    __syncthreads();

    // --- row sums via WMMA: D = A(16x4) * ones(4x16) + C, chained over K ---
    // f32 A fragment layout: lanes 0-15 hold (K=k0,k0+1), lanes 16-31 (K=k0+2,k0+3)
    {
        const int lane = tid & 31;
        const int wave = tid >> 5;
        const int row  = lane & 15;
        const int koff = (lane >> 4) * 2;
        const float* rp = &smem[row * LSTRIDE + wave * 128 + koff];
        v2f bones = {1.0f, 1.0f};
        v8f acc = {0.f, 0.f, 0.f, 0.f, 0.f, 0.f, 0.f, 0.f};
        #pragma unroll
        for (int kk = 0; kk < 32; ++kk) {   // this wave covers K = wave*128 .. +127
            v2f a = *(const v2f*)(rp + kk * 4);
            acc = __builtin_amdgcn_wmma_f32_16x16x4_f32(
                false, a, false, bones, (short)0, acc, false, false);
        }
        // every column of D equals the row sum; take column n==0 (lanes 0 and 16)
        if (row == 0) {
            const int mb = (lane >> 4) * 8;  // lane 0 -> M 0..7, lane 16 -> M 8..15
            #pragma unroll
            for (int v = 0; v < 8; ++v) s_part[wave][mb + v] = acc[v];
        }
    }
    __syncthreads();
    if (tid < ROWS) {   // deterministic fixed-order combine of 8 wave partials
        float s = 0.0f;
        #pragma unroll
        for (int w = 0; w < 8; ++w) s += s_part[w][tid];
        s_scale[tid] = rsqrtf(s);            // reference: mask / sqrt(rowsum)
    }
    __syncthreads();

    // --- normalize + coalesced store of the mask strip ---
    const size_t mbase = ((size_t)bh * SLEN + qbase) * (size_t)SLEN;
    #pragma unroll
    for (int k = 0; k < ROWS; ++k) {
        const float sc = s_scale[k];
        float4 v = *(const float4*)&smem[k * LSTRIDE + c0];
        v.x *= sc; v.y *= sc; v.z *= sc; v.w *= sc;
        *(float4*)&mask_out[mbase + (size_t)k * SLEN + c0] = v;
    }

    // --- rotary q for this strip: 16 queries x 64 dims ---
    {
        const int q  = tid >> 4;          // 0..15
        const int d0 = (tid & 15) * 4;    // 0..60
        const float idx = s_idxq[q];
        const float4 ang = *(const float4*)&angle[d0];
        float4 s, c;
        __sincosf(idx * ang.x, &s.x, &c.x);
        __sincosf(idx * ang.y, &s.y, &c.y);
        __sincosf(idx * ang.z, &s.z, &c.z);
        __sincosf(idx * ang.w, &s.w, &c.w);
        const size_t base = ((size_t)bh * SLEN + qbase + q) * DHEAD + d0;
        *(float4*)&sin_q[base] = s;
        *(float4*)&cos_q[base] = c;
    }
}

// sin_k/cos_k: [1024, 64]; 64 blocks x 256 threads, 4 dims per thread.
__global__ __launch_bounds__(256)
void rotary_k_kernel(const float* __restrict__ angle,
                     float* __restrict__ sin_k,
                     float* __restrict__ cos_k)
{
    const int t  = blockIdx.x * 256 + threadIdx.x;  // 0..16383
    const int q  = t >> 4;                          // 0..1023
    const int d0 = (t & 15) * 4;
    const float idx = (float)((q >> 5) + (q & 31)); // index_k = i + j
    const float4 ang = *(const float4*)&angle[d0];
    float4 s, c;
    __sincosf(idx * ang.x, &s.x, &c.x);
    __sincosf(idx * ang.y, &s.y, &c.y);
    __sincosf(idx * ang.z, &s.z, &c.z);
    __sincosf(idx * ang.w, &s.w, &c.w);
    const size_t base = (size_t)q * DHEAD + d0;
    *(float4*)&sin_k[base] = s;
    *(float4*)&cos_k[base] = c;
}

extern "C" void kernel_launch(void* const* d_in, const int* in_sizes, int n_in,
                              void* d_out, int out_size, void* d_ws, size_t ws_size,
                              hipStream_t stream) {
    // inputs: slen(scalar, unused), offsets[8,8,32,32,2], angle[64], decay[8]
    const float* offsets = (const float*)d_in[1];
    const float* angle   = (const float*)d_in[2];
    const float* decay   = (const float*)d_in[3];

    float* out = (float*)d_out;
    const size_t qelems = (size_t)NB * NH * SLEN * DHEAD;  // 4,194,304
    const size_t kelems = (size_t)SLEN * DHEAD;            //    65,536
    float* sq = out;
    float* cq = sq + qelems;
    float* sk = cq + qelems;
    float* ck = sk + kelems;
    float* mk = ck + kelems;

    deform_strip_kernel<<<NB * NH * (SLEN / ROWS), 256, 0, stream>>>(
        offsets, angle, decay, sq, cq, mk);
    rotary_k_kernel<<<(SLEN * DHEAD / 4) / 256, 256, 0, stream>>>(angle, sk, ck);
}